// NegativeSamplingLoss_16965120820078
// MI455X (gfx1250) — compile-verified
//
#include <hip/hip_runtime.h>
#include <hip/hip_bf16.h>
#include <stdint.h>

// ---------------------------------------------------------------------------
// NegativeSamplingLoss on MI455X (gfx1250, wave32).
// Streaming top-16-per-row + log-sigmoid reduction. Bandwidth bound (256 MB @
// 23.3 TB/s ~= 11us). Data path uses CDNA5 async global->LDS loads
// (global_load_async_to_lds_b128, ASYNCcnt) with double buffering.
// ---------------------------------------------------------------------------

#define TILE_COLS   512          // columns staged per tile (2 KB per row)
#define ROWS_PER_BLK 8           // one wave32 per row
#define NNEG        16

// Low 32 bits of a flat LDS pointer == LDS byte offset (ISA 10.2).
__device__ __forceinline__ unsigned lds_off_u32(const void* p) {
    return (unsigned)(unsigned long long)p;
}

// CDNA5 async copy: per-lane 16B global -> LDS, tracked by ASYNCcnt.
__device__ __forceinline__ void async_ld_b128(unsigned lds_byte_off, const void* gaddr) {
    asm volatile("global_load_async_to_lds_b128 %0, %1, off"
                 :: "v"(lds_byte_off), "v"(gaddr)
                 : "memory");
}
__device__ __forceinline__ void wait_async_le4() {
    asm volatile("s_wait_asynccnt 4" ::: "memory");
}
__device__ __forceinline__ void wait_async_0() {
    asm volatile("s_wait_asynccnt 0" ::: "memory");
}

// Numerically stable log-sigmoid: log_sigmoid(z) = min(z,0) - log1p(exp(-|z|))
__device__ __forceinline__ float lsig(float z) {
    return fminf(z, 0.0f) - log1pf(expf(-fabsf(z)));
}

__global__ void __launch_bounds__(256)
nsl_topk_rows_kernel(const float* __restrict__ x,
                     const int*   __restrict__ sel,
                     float*       __restrict__ ws,
                     int n)                       // n = rows = cols
{
    // [buf][wave][col] : 2 * 8 * 512 * 4B = 32 KB
    __shared__ float ldsx[2][ROWS_PER_BLK][TILE_COLS];

    const int lane = threadIdx.x & 31;
    const int wv   = threadIdx.x >> 5;
    const int row  = blockIdx.x * ROWS_PER_BLK + wv;
    if (row >= n) return;

    const float* gx   = x + (size_t)row * (size_t)n;
    const int    selr = sel[row];
    const int    NT   = n / TILE_COLS;

    const unsigned base0 = lds_off_u32(&ldsx[0][wv][0]);
    const unsigned base1 = lds_off_u32(&ldsx[1][wv][0]);

    // Per-lane sorted (descending) top-16; 0-init matches x*mask semantics
    // (masked entries contribute exactly 0.0f).
    float t[NNEG];
#pragma unroll
    for (int i = 0; i < NNEG; ++i) t[i] = 0.0f;

    // Branchless sorted insert (only called when val > t[15]).
    auto insert16 = [&](float val) {
#pragma unroll
        for (int k = NNEG - 1; k >= 1; --k) {
            float a = t[k - 1], b = t[k];
            t[k] = (val > a) ? a : ((val > b) ? val : b);
        }
        t[0] = fmaxf(t[0], val);
    };
    auto consider = [&](float v, int sj) {
        float val = (sj != selr) ? v : 0.0f;
        if (val > t[NNEG - 1]) insert16(val);
    };

    // Issue a 512-col tile: 4 x b128 per lane (each covers 32 lanes * 16B = 512B).
    auto issue_tile = [&](int tt, unsigned base) {
        const float* gsrc = gx + (size_t)tt * TILE_COLS;
#pragma unroll
        for (int k = 0; k < 4; ++k) {
            const float* ga = gsrc + k * 128 + lane * 4;
            unsigned     lo = base + (unsigned)((k * 128 + lane * 4) * 4);
            async_ld_b128(lo, (const void*)ga);
        }
    };

    // Software pipeline: prefetch tile 0, then overlap issue(t+1) with consume(t).
    issue_tile(0, base0);
    for (int tt = 0; tt < NT; ++tt) {
        const int      buf  = tt & 1;
        const unsigned bcur = buf ? base1 : base0;
        (void)bcur;
        if (tt + 1 < NT) {
            issue_tile(tt + 1, (buf ^ 1) ? base1 : base0);
            wait_async_le4();              // oldest 4 (tile tt) complete, in order
        } else {
            wait_async_0();
        }

        // Consume: lane owns 16 contiguous floats of this tile.
        const float4* lp = (const float4*)&ldsx[buf][wv][lane * 16];
        const int4*   sp = (const int4*)(sel + (size_t)tt * TILE_COLS + lane * 16);
#pragma unroll
        for (int q = 0; q < 4; ++q) {
            float4 xv = lp[q];
            int4   sv = sp[q];
            consider(xv.x, sv.x);
            consider(xv.y, sv.y);
            consider(xv.z, sv.z);
            consider(xv.w, sv.w);
        }
    }

    // Merge 32 lanes * top-16 -> exact row top-16.
    // Dump sorted per-lane lists to LDS (wave-private slot), then 16 rounds of
    // wave-max selection via shfl_xor butterflies (wave32).
    float* dump = &ldsx[0][wv][0];
#pragma unroll
    for (int i = 0; i < NNEG; ++i) dump[lane * NNEG + i] = t[i];

    int   head   = 0;
    float negsum = 0.0f;
    for (int k = 0; k < NNEG; ++k) {
        float cand;
        if (head < NNEG) cand = dump[lane * NNEG + head];
        else             cand = -3.0e38f;
        float m = cand;
#pragma unroll
        for (int off = 16; off >= 1; off >>= 1)
            m = fmaxf(m, __shfl_xor(m, off, 32));
        unsigned long long bal = __ballot(cand == m);
        int leader = __ffsll((long long)bal) - 1;
        if (lane == leader) head++;
        negsum += lsig(-m);                // wave-uniform
    }

    if (lane == 0) {
        const float invN  = 1.0f / (float)n;
        const float invNK = 1.0f / ((float)n * (float)NNEG);
        float diag = gx[row];              // x[row, row]
        ws[row] = -lsig(diag) * invN - negsum * invNK;
    }
}

__global__ void __launch_bounds__(256)
nsl_reduce_kernel(const float* __restrict__ ws, float* __restrict__ out, int n)
{
    __shared__ float s[256];
    float acc = 0.0f;
    for (int i = threadIdx.x; i < n; i += 256) acc += ws[i];   // fixed order: deterministic
    s[threadIdx.x] = acc;
    __syncthreads();
#pragma unroll
    for (int off = 128; off > 0; off >>= 1) {
        if ((int)threadIdx.x < off) s[threadIdx.x] += s[threadIdx.x + off];
        __syncthreads();
    }
    if (threadIdx.x == 0) out[0] = s[0];
}

extern "C" void kernel_launch(void* const* d_in, const int* in_sizes, int n_in,
                              void* d_out, int out_size, void* d_ws, size_t ws_size,
                              hipStream_t stream) {
    const float* x   = (const float*)d_in[0];
    const int*   sel = (const int*)d_in[1];
    float*       out = (float*)d_out;
    float*       ws  = (float*)d_ws;
    const int    n   = in_sizes[1];        // 8192 (square matrix)

    dim3 grid((n + ROWS_PER_BLK - 1) / ROWS_PER_BLK);
    dim3 block(256);
    nsl_topk_rows_kernel<<<grid, block, 0, stream>>>(x, sel, ws, n);
    nsl_reduce_kernel<<<1, 256, 0, stream>>>(ws, out, n);
}